// MHTrainingLoss_90142773608452
// MI455X (gfx1250) — compile-verified
//
#include <hip/hip_runtime.h>
#include <math.h>

#define VOCAB_SZ 2048
#define CHORD_SZ 60
#define REP_WINDOW 8

typedef __attribute__((ext_vector_type(2))) float v2f;
typedef __attribute__((ext_vector_type(8))) float v8f;

__device__ __forceinline__ float warp_max(float v) {
#pragma unroll
  for (int m = 16; m >= 1; m >>= 1) v = fmaxf(v, __shfl_xor(v, m, 32));
  return v;
}
__device__ __forceinline__ float warp_sum(float v) {
#pragma unroll
  for (int m = 16; m >= 1; m >>= 1) v += __shfl_xor(v, m, 32);
  return v;
}

// ---------------------------------------------------------------------------
// Main CE: one 256-thread block (8 waves) per row of 2048 logits.
// Each thread holds 8 floats in registers (two coalesced b128 loads),
// max via shfl+LDS, exp on TRANS pipe, sum via chained V_WMMA_F32_16X16X4_F32
// against a ones matrix (64 f32 per WMMA, f32 C accumulation -> exact).
// ---------------------------------------------------------------------------
__global__ __launch_bounds__(256) void ce_main_kernel(
    const float* __restrict__ logits, const int* __restrict__ tgt,
    float* __restrict__ out_nll, float* __restrict__ out_valid) {
  const int row  = blockIdx.x;
  const int tid  = threadIdx.x;
  const int lane = tid & 31;
  const int wave = tid >> 5;
  __shared__ float s_red[8];
  __shared__ float s_xt;

  const float4* r4 = (const float4*)(logits + (size_t)row * VOCAB_SZ);
  float4 va = r4[tid];        // elements [tid*4, tid*4+4)
  float4 vb = r4[256 + tid];  // elements [1024 + tid*4, ...)
  const int t = tgt[row];
  if (tid == 0) s_xt = 0.0f;

  float e[8] = {va.x, va.y, va.z, va.w, vb.x, vb.y, vb.z, vb.w};
  float tmax = e[0];
#pragma unroll
  for (int i = 1; i < 8; ++i) tmax = fmaxf(tmax, e[i]);
  tmax = warp_max(tmax);
  if (lane == 0) s_red[wave] = tmax;
  __syncthreads();
  float m = s_red[0];
#pragma unroll
  for (int i = 1; i < 8; ++i) m = fmaxf(m, s_red[i]);

  // capture x[target] (at most one thread/component matches)
  const int base0 = tid * 4;
#pragma unroll
  for (int c = 0; c < 4; ++c) {
    if (base0 + c == t)        s_xt = e[c];
    if (1024 + base0 + c == t) s_xt = e[4 + c];
  }
#pragma unroll
  for (int i = 0; i < 8; ++i) e[i] = __expf(e[i] - m);
  __syncthreads();  // everyone done reading s_red before reuse

  float wsum;
#if defined(__HIP_DEVICE_COMPILE__) && __has_builtin(__builtin_amdgcn_wmma_f32_16x16x4_f32)
  {
    // D = A(16x4) * ones(4x16) + C : each WMMA folds 64 f32 into C; chain 4x.
    v8f cacc = {};
    v2f ones; ones[0] = 1.0f; ones[1] = 1.0f;
#pragma unroll
    for (int i = 0; i < 4; ++i) {
      v2f av; av[0] = e[2 * i]; av[1] = e[2 * i + 1];
      cacc = __builtin_amdgcn_wmma_f32_16x16x4_f32(
          false, av, false, ones, (short)0, cacc, false, false);
    }
    // lane holds 8 C rows of its column; columns are identical, xor16 adds
    // the complementary 8 rows -> every lane has the full wave sum.
    float s = 0.0f;
#pragma unroll
    for (int i = 0; i < 8; ++i) s += cacc[i];
    s += __shfl_xor(s, 16, 32);
    wsum = s;
  }
#else
  {
    float s = 0.0f;
#pragma unroll
    for (int i = 0; i < 8; ++i) s += e[i];
    wsum = warp_sum(s);
  }
#endif
  if (lane == 0) s_red[wave] = wsum;
  __syncthreads();
  if (tid == 0) {
    float tot = 0.0f;
#pragma unroll
    for (int i = 0; i < 8; ++i) tot += s_red[i];
    const bool valid = (t != -100);
    const float logZ = m + logf(tot);
    out_nll[row]   = valid ? (logZ - s_xt) : 0.0f;
    out_valid[row] = valid ? 1.0f : 0.0f;
  }
}

// ---------------------------------------------------------------------------
// Chord CE: one wave per row of 60 logits (lane covers i and i+32).
// ---------------------------------------------------------------------------
__global__ __launch_bounds__(256) void ce_chord_kernel(
    const float* __restrict__ x, const int* __restrict__ tgt,
    float* __restrict__ out_nll, float* __restrict__ out_valid, int nrows) {
  const int wave = threadIdx.x >> 5;
  const int lane = threadIdx.x & 31;
  const int row  = blockIdx.x * 8 + wave;
  if (row >= nrows) return;
  const float* r = x + (size_t)row * CHORD_SZ;
  float v1 = r[lane];
  float v2 = (lane + 32 < CHORD_SZ) ? r[lane + 32] : -__builtin_inff();
  float m = warp_max(fmaxf(v1, v2));
  float s = __expf(v1 - m) + ((lane + 32 < CHORD_SZ) ? __expf(v2 - m) : 0.0f);
  s = warp_sum(s);
  const int t = tgt[row];
  const bool valid = (t != -1000000000);
  const int tt = valid ? t : 0;
  float xt = (tt < 32) ? __shfl(v1, tt & 31, 32) : __shfl(v2, (tt - 32) & 31, 32);
  if (lane == 0) {
    out_nll[row]   = valid ? (m + logf(s) - xt) : 0.0f;
    out_valid[row] = valid ? 1.0f : 0.0f;
  }
}

// ---------------------------------------------------------------------------
// BCE-with-logits: grid-stride float4 stream, per-block partial sums.
// ---------------------------------------------------------------------------
__global__ __launch_bounds__(256) void bce_scale_kernel(
    const float* __restrict__ x, const float* __restrict__ z,
    float* __restrict__ out_part, int n4) {
  const int tid = threadIdx.x;
  float acc = 0.0f;
  const float4* x4 = (const float4*)x;
  const float4* z4 = (const float4*)z;
  for (int i = blockIdx.x * blockDim.x + tid; i < n4;
       i += gridDim.x * blockDim.x) {
    float4 a = x4[i], b = z4[i];
    const float xs[4] = {a.x, a.y, a.z, a.w};
    const float zs[4] = {b.x, b.y, b.z, b.w};
#pragma unroll
    for (int c = 0; c < 4; ++c) {
      float xv = xs[c];
      acc += fmaxf(xv, 0.0f) - xv * zs[c] + log1pf(__expf(-fabsf(xv)));
    }
  }
  __shared__ float s_red[8];
  float w = warp_sum(acc);
  if ((tid & 31) == 0) s_red[tid >> 5] = w;
  __syncthreads();
  if (tid == 0) {
    float tot = 0.0f;
#pragma unroll
    for (int i = 0; i < 8; ++i) tot += s_red[i];
    out_part[blockIdx.x] = tot;
  }
}

// ---------------------------------------------------------------------------
// Finalize: deterministic fixed-order tree reduction of all partials,
// closed-form repetition constant, weighted combine -> scalar.
// ---------------------------------------------------------------------------
__global__ __launch_bounds__(256) void finalize_kernel(
    const float* __restrict__ ws, int nrows, int nsb, int nscale, int batch,
    float* __restrict__ out) {
  __shared__ float sh[256];
  const int tid = threadIdx.x;
  float sums[5];
  const float* ptrs[5] = {ws, ws + nrows, ws + 2 * nrows, ws + 3 * nrows,
                          ws + 4 * nrows};
  const int lens[5] = {nrows, nrows, nrows, nrows, nsb};
  for (int k = 0; k < 5; ++k) {
    float acc = 0.0f;
    for (int i = tid; i < lens[k]; i += 256) acc += ptrs[k][i];
    sh[tid] = acc;
    __syncthreads();
    for (int s = 128; s > 0; s >>= 1) {
      if (tid < s) sh[tid] += sh[tid + s];
      __syncthreads();
    }
    sums[k] = sh[0];
    __syncthreads();
  }
  if (tid == 0) {
    const float main_mean  = sums[0] / fmaxf(sums[1], 1.0f);
    const float chord_mean = sums[2] / fmaxf(sums[3], 1.0f);
    const float scale_mean = sums[4] / (float)nscale;
    // rep_loss = 0.5 * mean(counts); sum_v counts[b,p,v] == min(p, W) exactly
    const int S = nrows / batch;
    const double cnt = (double)(REP_WINDOW * (REP_WINDOW - 1) / 2) +
                       (double)REP_WINDOW * (double)(S - REP_WINDOW);
    const float rep = (float)(0.5 * cnt / ((double)S * (double)VOCAB_SZ));
    out[0] = main_mean + 0.05f * rep + 0.2f * chord_mean + 0.1f * scale_mean;
  }
}

extern "C" void kernel_launch(void* const* d_in, const int* in_sizes, int n_in,
                              void* d_out, int out_size, void* d_ws,
                              size_t ws_size, hipStream_t stream) {
  const float* logits        = (const float*)d_in[0];
  const float* chord_logits  = (const float*)d_in[1];
  const float* scale_logits  = (const float*)d_in[2];
  const float* scale_targets = (const float*)d_in[3];
  const int*   target_ids    = (const int*)d_in[4];
  // d_in[5] = key_ids (unused by the reference loss)
  const int*   chord_targets = (const int*)d_in[6];

  const int nrows  = in_sizes[4];  // B*S = 8192
  const int batch  = in_sizes[5];  // B = 4
  const int nscale = in_sizes[2];  // B*S*12
  const int NSB    = 96;           // scale-kernel blocks

  float* ws = (float*)d_ws;
  // ws layout: [0,NR) main nll | [NR,2NR) main valid |
  //            [2NR,3NR) chord nll | [3NR,4NR) chord valid | [4NR,4NR+NSB) scale
  ce_main_kernel<<<nrows, 256, 0, stream>>>(logits, target_ids, ws, ws + nrows);
  ce_chord_kernel<<<(nrows + 7) / 8, 256, 0, stream>>>(
      chord_logits, chord_targets, ws + 2 * nrows, ws + 3 * nrows, nrows);
  bce_scale_kernel<<<NSB, 256, 0, stream>>>(scale_logits, scale_targets,
                                            ws + 4 * nrows, nscale / 4);
  finalize_kernel<<<1, 256, 0, stream>>>(ws, nrows, NSB, nscale, batch,
                                         (float*)d_out);
}